// BallQLoss_15762529976906
// MI455X (gfx1250) — compile-verified
//
#include <hip/hip_runtime.h>

// BallQLoss for MI455X (gfx1250, wave32).
// Gram-matrix pairwise distances via V_WMMA_F32_16X16X4_F32, ordered first-K
// ball-query selection via wave ballots + per-lane bit scans, deterministic
// two-stage reduction. Problem is L2-resident (~400KB), so the only real work
// is the B*N^2 distance evals -> WMMA, and the serial first-K scan -> 1 query
// per lane with early exit.

typedef __attribute__((ext_vector_type(2))) float v2f;
typedef __attribute__((ext_vector_type(8))) float v8f;

#define KNN 16
#define RADIUS2 0.25f

__global__ __launch_bounds__(32)
void ballq_loss_kernel(const float* __restrict__ pc,
                       const float* __restrict__ flow,
                       float* __restrict__ partial,
                       int B, int N) {
    const int lane = threadIdx.x;                 // 0..31 (wave32)
    const int tilesPerBatch = N / 32;
    const int b = blockIdx.x / tilesPerBatch;
    const int rowbase = (blockIdx.x % tilesPerBatch) * 32;  // queries [rowbase, rowbase+32)
    const float* __restrict__ pcB   = pc   + (size_t)b * N * 3;
    const float* __restrict__ flowB = flow + (size_t)b * N * 3;

    const int  mA  = lane & 15;
    const bool hiA = (lane >= 16);

    // ---- A tiles: 16x4 f32 (rows = queries, K = xyz + zero pad).
    // ISA 7.12.2 layout: lanes 0-15 row m=lane {V0=K0, V1=K1};
    //                    lanes 16-31 row m=lane-16 {V0=K2, V1=K3}.
    v2f A0, A1;
    {
        int m0 = rowbase + mA;
        float x0 = pcB[m0 * 3 + 0], y0 = pcB[m0 * 3 + 1], z0 = pcB[m0 * 3 + 2];
        A0.x = hiA ? z0 : x0;
        A0.y = hiA ? 0.0f : y0;
        int m1 = rowbase + 16 + mA;
        float x1 = pcB[m1 * 3 + 0], y1 = pcB[m1 * 3 + 1], z1 = pcB[m1 * 3 + 2];
        A1.x = hiA ? z1 : x1;
        A1.y = hiA ? 0.0f : y1;
    }

    // ---- |pi|^2 per C/D element position: C vgpr v, lanes 0-15 -> row v,
    // lanes 16-31 -> row v+8 (ISA C/D 16x16 f32 layout).
    float sqr0[8], sqr1[8];
    {
        const int roff = hiA ? 8 : 0;
#pragma unroll
        for (int v = 0; v < 8; ++v) {
            int m = rowbase + v + roff;
            float x = pcB[m * 3 + 0], y = pcB[m * 3 + 1], z = pcB[m * 3 + 2];
            sqr0[v] = x * x + y * y + z * z;
            m += 16;
            x = pcB[m * 3 + 0]; y = pcB[m * 3 + 1]; z = pcB[m * 3 + 2];
            sqr1[v] = x * x + y * y + z * z;
        }
    }

    // ---- this lane's query flow (lane r owns query rowbase+r)
    const int q = rowbase + lane;
    const float fqx = flowB[q * 3 + 0];
    const float fqy = flowB[q * 3 + 1];
    const float fqz = flowB[q * 3 + 2];

    __shared__ unsigned smask[32];

    unsigned cnt = 0;
    float acc = 0.0f;
    float firstL1 = 0.0f;

    const int numTiles = N / 16;
    for (int jt = 0; jt < numTiles; ++jt) {
        const int jbase = jt * 16;

        // ---- B tile: 4x16 f32 (rows = K, cols = candidate points).
        // Mirrors A layout: lanes 0-15 col n=lane {V0=K0,V1=K1};
        //                   lanes 16-31 col n=lane-16 {V0=K2,V1=K3(pad)}.
        const int jc = jbase + mA;
        const float bx = pcB[jc * 3 + 0];
        const float by = pcB[jc * 3 + 1];
        const float bz = pcB[jc * 3 + 2];
        const float sqj = bx * bx + by * by + bz * bz;
        v2f Bv;
        Bv.x = hiA ? bz : bx;
        Bv.y = hiA ? 0.0f : by;

        v8f czero = {};
        // dot(i,j) tiles: D = A x B + 0
        v8f d0 = __builtin_amdgcn_wmma_f32_16x16x4_f32(
            false, A0, false, Bv, (short)0, czero, false, false);
        v8f d1 = __builtin_amdgcn_wmma_f32_16x16x4_f32(
            false, A1, false, Bv, (short)0, czero, false, false);

        // ---- in-ball masks per row; ballot bits 0-15 = row v (cols 0-15),
        // bits 16-31 = row v+8 (cols 0-15).
#pragma unroll
        for (int v = 0; v < 8; ++v) {
            float dd0 = sqr0[v] + sqj - 2.0f * d0[v];
            float dd1 = sqr1[v] + sqj - 2.0f * d1[v];
            unsigned m0 = (unsigned)__ballot(dd0 < RADIUS2);
            unsigned m1 = (unsigned)__ballot(dd1 < RADIUS2);
            if (lane == 0) {
                smask[v]      = m0 & 0xFFFFu;
                smask[v + 8]  = m0 >> 16;
                smask[16 + v] = m1 & 0xFFFFu;
                smask[24 + v] = m1 >> 16;
            }
        }
        __syncthreads();
        unsigned mask = smask[lane];
        __syncthreads();

        // ---- ordered first-K scan for this lane's query
        if (cnt < KNN) {
            while (mask) {
                const int bit = __builtin_ctz(mask);
                mask &= mask - 1;
                const int j = jbase + bit;
                const float l1 = fabsf(fqx - flowB[j * 3 + 0]) +
                                 fabsf(fqy - flowB[j * 3 + 1]) +
                                 fabsf(fqz - flowB[j * 3 + 2]);
                if (cnt == 0) firstL1 = l1;
                acc += l1;
                if (++cnt >= KNN) break;
            }
        }

        // uniform early exit once every query in the wave has K neighbors
        if (__all(cnt >= KNN)) break;
    }

    // pad empty slots with first neighbor's contribution (self always in-ball)
    if (cnt > 0 && cnt < KNN) acc += (float)(KNN - cnt) * firstL1;

    // wave reduction -> per-block partial
#pragma unroll
    for (int off = 16; off > 0; off >>= 1) acc += __shfl_down(acc, off);
    if (lane == 0) partial[blockIdx.x] = acc;
}

__global__ __launch_bounds__(256)
void ballq_reduce_kernel(const float* __restrict__ partial, int n,
                         float* __restrict__ out, float invCount) {
    __shared__ float s[256];
    float a = 0.0f;
    for (int i = threadIdx.x; i < n; i += 256) a += partial[i];
    s[threadIdx.x] = a;
    __syncthreads();
#pragma unroll
    for (int off = 128; off > 0; off >>= 1) {
        if ((int)threadIdx.x < off) s[threadIdx.x] += s[threadIdx.x + off];
        __syncthreads();
    }
    if (threadIdx.x == 0) out[0] = s[0] * invCount;
}

extern "C" void kernel_launch(void* const* d_in, const int* in_sizes, int n_in,
                              void* d_out, int out_size, void* d_ws, size_t ws_size,
                              hipStream_t stream) {
    const float* pc   = (const float*)d_in[0];
    const float* flow = (const float*)d_in[1];
    float* out = (float*)d_out;

    const int B = 2;
    const int N = in_sizes[0] / (B * 3);   // 8192
    const int nBlocks = B * (N / 32);      // 512

    float* partial = (float*)d_ws;

    ballq_loss_kernel<<<nBlocks, 32, 0, stream>>>(pc, flow, partial, B, N);

    const float invCount = 1.0f / ((float)B * (float)N * (float)KNN);
    ballq_reduce_kernel<<<1, 256, 0, stream>>>(partial, nBlocks, out, invCount);
}